// CostVolumeLayer3D_51153060496183
// MI455X (gfx1250) — compile-verified
//
#include <hip/hip_runtime.h>

typedef __attribute__((ext_vector_type(2))) float v2f;
typedef __attribute__((ext_vector_type(8))) float v8f;
typedef unsigned int v4u __attribute__((ext_vector_type(4)));
typedef int v8i __attribute__((ext_vector_type(8)));
typedef int v4i __attribute__((ext_vector_type(4)));

// Problem constants (match reference setup_inputs exactly)
#define BDIM 2
#define CDIM 64
#define DDIM 32
#define HDIM 64
#define WDIM 64
#define NCH  125
#define CSTR (DDIM*HDIM*WDIM)    // 131072 : channel stride in x1/x2
#define DSTR (HDIM*WDIM)         // 4096
#define BSTR ((size_t)CDIM*CSTR) // 8388608 : batch stride in x1/x2

// LDS layout (in floats)
#define X1_PITCH 80
#define X2_PITCH 80
#define OUT_PITCH 66
#define LDS_X1   0
#define LDS_X2   (LDS_X1 + CDIM*X1_PITCH)      // 5120
#define LDS_OUT  (LDS_X2 + CDIM*X2_PITCH)      // 10240
#define LDS_CORR (LDS_OUT + NCH*OUT_PITCH)     // 18490
#define LDS_TOTAL (LDS_CORR + 8*16*17)         // 20666 floats = 82664 B

#define HAVE_TDM __has_builtin(__builtin_amdgcn_tensor_load_to_lds)

#if HAVE_TDM
// TDM descriptor for one 64(c) x 64(w) fp32 tile, global row stride 131072
// elements, written to LDS with pitch 80 dwords (64 data + 16 pad skipped).
// Field packing per CDNA5 ISA. This toolchain
// exposes the 6-arg builtin: (g0 v4u, g1 v8i, g2 v4i, g3 v4i, v8i, cpol);
// groups 2/3 and the trailing operand are only needed for >=3D tensors.
__device__ __forceinline__ void tdm_tile_load(unsigned lds_byte_addr,
                                              const float* gptr)
{
    const unsigned long long ga = (unsigned long long)(uintptr_t)gptr;
    v4u g0;
    g0.x = 1u;                                   // count=1, user mode
    g0.y = lds_byte_addr;                        // lds_addr [63:32]
    g0.z = (unsigned)(ga & 0xffffffffu);         // global_addr [95:64]
    g0.w = (unsigned)((ga >> 32) & 0x01ffffffu)  // global_addr [120:96]
         | (2u << 30);                           // type=2 ("image") [127:126]

    v8i g1;
    g1[0] = (2 << 16)        // data_size = 2 (4 bytes)
          | (1 << 20)        // pad_enable
          | (5 << 22)        // pad_interval: code 5 = every 64 dwords
          | (15 << 25);      // pad_amount:   code 15 = 16 dwords
    g1[1] = (64 << 16);      // tensor_dim0 = 64   (bits 79:48)
    g1[2] = (64 << 16);      // tensor_dim1 = 64   (bits 111:80)
    g1[3] = (64 << 16);      // tile_dim0   = 64   (bits 127:112)
    g1[4] = 64;              // tile_dim1   = 64, tile_dim2 = 0
    g1[5] = CSTR;            // tensor_dim0_stride = 131072 (bits 207:160)
    g1[6] = 0;               // stride hi bits, tensor_dim1_stride = 0
    g1[7] = 0;
    const v4i gz4 = {0, 0, 0, 0};                // groups 2/3 unused (2D)
    const v8i gz8 = {0, 0, 0, 0, 0, 0, 0, 0};    // trailing operand unused
    __builtin_amdgcn_tensor_load_to_lds(g0, g1, gz4, gz4, gz8, 0);
}
#endif

__global__ __launch_bounds__(256)
void costvol3d_kernel(const float* __restrict__ x1,
                      const float* __restrict__ x2,
                      float* __restrict__ out)
{
    extern __shared__ float smem[];
    float* lx1   = smem + LDS_X1;
    float* lx2   = smem + LDS_X2;
    float* lout  = smem + LDS_OUT;
    float* lcorr = smem + LDS_CORR;

    const int tid  = threadIdx.x;
    const int wave = tid >> 5;
    const int lane = tid & 31;

    const int row = blockIdx.x;      // 0 .. B*D*H-1 = 4095
    const int b   = row >> 11;       // / (32*64)
    const int d0  = (row >> 6) & 31;
    const int h0  = row & 63;

    const float* p1 = x1 + (size_t)b*BSTR + (size_t)d0*DSTR + (size_t)h0*WDIM;

    // ---------------- Phase 1: stage x1 row, zero lout + x2 halo slots ----
#if HAVE_TDM
    if (wave == 0) tdm_tile_load((unsigned)(LDS_X1 * 4), p1);
#else
    for (int k = tid; k < CDIM*16; k += 256) {      // 1024 float4 chunks
        const int c  = k >> 4;
        const int w4 = (k & 15) << 2;
        const float4 v = *reinterpret_cast<const float4*>(p1 + (size_t)c*CSTR + w4);
        *reinterpret_cast<float4*>(&lx1[c*X1_PITCH + w4]) = v;
    }
#endif
    for (int i = tid; i < NCH*OUT_PITCH; i += 256) lout[i] = 0.0f;
    // x2 band halo: slots 0..3 (q=-4..-1) and 68..79 (q>=64) stay zero; the
    // TDM pad feature skips them on every refill.
    {
        const float4 z = make_float4(0.f, 0.f, 0.f, 0.f);
        for (int k = tid; k < CDIM*4; k += 256) {
            const int c = k >> 2, j = k & 3;
            const int s = (j == 0) ? 0 : (64 + 4*j);   // 0 | 68 | 72 | 76
            *reinterpret_cast<float4*>(&lx2[c*X2_PITCH + s]) = z;
        }
    }
#if HAVE_TDM
    if (wave == 0) __builtin_amdgcn_s_wait_tensorcnt(0);
#endif
    __syncthreads();

    // ---------------- Phase 2: W-banded WMMA group (di=+2, dj in [-2,2]) --
    // For each dh: TDM-stage x2 row (d0-dh, h0-2) interior (slot = q+4),
    // then 6 diagonal 16x16x64 fp32 WMMA tiles cover all 5 W-diagonals.
    for (int dhi = 0; dhi < 5; ++dhi) {
        const int dsrc = d0 - (dhi - 2);
        const int hsrc = h0 - 2;
        const bool rv = (dsrc >= 0) && (dsrc < DDIM) && (hsrc >= 0);
        const float* p2 = x2 + (size_t)b*BSTR + (size_t)dsrc*DSTR + (size_t)hsrc*WDIM;

#if HAVE_TDM
        if (rv) {
            if (wave == 0) {
                tdm_tile_load((unsigned)(LDS_X2 * 4 + 16), p2);  // slot 4
                __builtin_amdgcn_s_wait_tensorcnt(0);
            }
        } else {
            // OOB source row: interior must be zero
            const float4 z = make_float4(0.f, 0.f, 0.f, 0.f);
            for (int k = tid; k < CDIM*16; k += 256) {
                const int c  = k >> 4;
                const int s4 = 4 + ((k & 15) << 2);
                *reinterpret_cast<float4*>(&lx2[c*X2_PITCH + s4]) = z;
            }
        }
#else
        for (int k = tid; k < CDIM*20; k += 256) {
            const int c  = k / 20;
            const int q4 = (k % 20) << 2;            // slot base
            float e[4] = {0.f, 0.f, 0.f, 0.f};
            if (rv) {
                const float* pc = p2 + (size_t)c*CSTR;
                #pragma unroll
                for (int t = 0; t < 4; ++t) {
                    const int q = q4 + t - 4;        // slot = q + 4
                    if (q >= 0 && q < WDIM) e[t] = pc[q];
                }
            }
            *reinterpret_cast<float4*>(&lx2[c*X2_PITCH + q4]) =
                make_float4(e[0], e[1], e[2], e[3]);
        }
#endif
        // warm L2 for the next dh band row
        if (dhi < 4) {
            const int dn = dsrc - 1;
            if (dn >= 0 && dn < DDIM && hsrc >= 0) {
                const float* pn = x2 + (size_t)b*BSTR + (size_t)dn*DSTR + (size_t)hsrc*WDIM;
                const int c = tid >> 2, off = (tid & 3) << 4;
                __builtin_prefetch(pn + (size_t)c*CSTR + off, 0, 1);
            }
        }
        __syncthreads();

        if (wave < 6) {
            const int w0   = wave * 12;        // output tile base (12 useful)
            const int half = lane >> 4;
            const int l    = lane & 15;

            v8f acc = {0.f,0.f,0.f,0.f,0.f,0.f,0.f,0.f};
            #pragma unroll
            for (int c0 = 0; c0 < CDIM; c0 += 4) {
                const int ca = c0 + 2*half;    // K pair for this half-wave
                v2f a, bb;
                // A (16x4): M = q-window index, q = w0-2+m -> slot w0+m+2
                a[0]  = lx2[(ca+0)*X2_PITCH + w0 + l + 2];
                a[1]  = lx2[(ca+1)*X2_PITCH + w0 + l + 2];
                // B (4x16): N = output position, w = w0+n
                bb[0] = lx1[(ca+0)*X1_PITCH + w0 + l];
                bb[1] = lx1[(ca+1)*X1_PITCH + w0 + l];
                acc = __builtin_amdgcn_wmma_f32_16x16x4_f32(
                        false, a, false, bb, (short)0, acc, false, false);
            }

            // spill corr tile to per-wave LDS scratch (C/D layout)
            float* cw = lcorr + wave * 16 * 17;
            #pragma unroll
            for (int r = 0; r < 8; ++r)
                cw[(r + 8*half)*17 + l] = acc[r];

            // extract the 5 diagonals: out w = w0+u uses m = u - dj + 2
            const int nu = (w0 == 60) ? 4 : 12;
            for (int p = lane; p < 5*12; p += 32) {
                const int e = p / 12;          // e = dj + 2
                const int u = p % 12;
                if (u < nu) {
                    const int m  = u - e + 4;
                    const int ch = (5*e + dhi + 123) % 125; // (10+5*dj+dh) mod 125
                    lout[ch*OUT_PITCH + w0 + u] = cw[m*17 + u];
                }
            }
        }
        __syncthreads();
    }

    // ---------------- Phase 3: single-diagonal group (dj=-2, di in [-2,1]) -
    // out[ch][w] = sum_c x1[c,w] * x2[c, d0-dh, h0-di, w+2]; x2 straight
    // from the 192MB L2 (coalesced: lanes map to consecutive w).
    for (int idx = tid; idx < 20*WDIM; idx += 256) {
        const int chid = idx >> 6;
        const int w    = idx & 63;
        const int dii  = chid / 5;             // di = dii - 2
        const int dhi  = chid % 5;             // dh = dhi - 2
        const int dsrc = d0 - (dhi - 2);
        const int hsrc = h0 - (dii - 2);
        float s = 0.0f;
        if (dsrc >= 0 && dsrc < DDIM && hsrc >= 0 && hsrc < HDIM && w < WDIM - 2) {
            const float* p2 = x2 + (size_t)b*BSTR + (size_t)dsrc*DSTR
                                 + (size_t)hsrc*WDIM + (w + 2);
            const float* pl = lx1 + w;
            #pragma unroll 8
            for (int c = 0; c < CDIM; ++c)
                s += pl[c*X1_PITCH] * p2[(size_t)c*CSTR];
        }
        const int ch = 5*dii + dhi + 103;      // (5*(di-2)+dh) mod 125
        lout[ch*OUT_PITCH + w] = s;
    }
    __syncthreads();

    // ---------------- Phase 4: coalesced store of all 125 channels --------
    {
        const float scale = 1.0f / 125.0f;
        float* po = out + (size_t)b*NCH*CSTR + (size_t)d0*DSTR + (size_t)h0*WDIM;
        for (int i = tid; i < NCH*WDIM; i += 256) {
            const int ch = i >> 6;
            const int w  = i & 63;
            po[(size_t)ch*CSTR + w] = lout[ch*OUT_PITCH + w] * scale;
        }
    }
}

extern "C" void kernel_launch(void* const* d_in, const int* in_sizes, int n_in,
                              void* d_out, int out_size, void* d_ws, size_t ws_size,
                              hipStream_t stream) {
    (void)in_sizes; (void)n_in; (void)d_ws; (void)ws_size; (void)out_size;
    const float* x1 = (const float*)d_in[0];
    const float* x2 = (const float*)d_in[1];
    float* out = (float*)d_out;

    dim3 grid(BDIM * DDIM * HDIM);   // 4096 blocks: one per (b, d, h) row
    dim3 block(256);                 // 8 wave32 waves
    size_t smem = (size_t)LDS_TOTAL * sizeof(float);
    hipLaunchKernelGGL(costvol3d_kernel, grid, block, smem, stream, x1, x2, out);
}